// DifferentiableTransformer_72189810311706
// MI455X (gfx1250) — compile-verified
//
#include <hip/hip_runtime.h>

// ---------------------------------------------------------------------------
// Atom-density dilation (scatter-add of 13^3 boxes into a periodic 200^3 grid)
// MI455X / gfx1250. Atomic/scatter-bound workload: no GEMM -> no WMMA.
// CDNA5 path: Tensor Data Mover (tensor_load_to_lds) stages each atom's
// radial-density row into LDS once per workgroup; s_wait_tensorcnt syncs it.
// ---------------------------------------------------------------------------

typedef unsigned int v4u __attribute__((ext_vector_type(4)));
typedef int          v4i __attribute__((ext_vector_type(4)));
typedef int          v8i __attribute__((ext_vector_type(8)));

#define BOX   13
#define NVOX  (BOX * BOX * BOX)      // 2197
#define TPB   256                    // 8 wave32s per workgroup
#define LDS_ROW_CAP 512              // > nbins (300)

__global__ __launch_bounds__(TPB) void atom_splat_kernel(
    const float* __restrict__ coords,          // (N,3) grid units
    const unsigned char* __restrict__ active,  // (N) bool, 1 byte each
    const float* __restrict__ occ,             // (N)
    const float* __restrict__ lmax,            // (3)
    const float* __restrict__ rad,             // (N,nbins) radial densities
    const float* __restrict__ rstep_p,         // scalar
    const float* __restrict__ rmax_p,          // scalar
    const float* __restrict__ G,               // (3,3) grid_to_cartesian, row-major
    float* __restrict__ out,                   // (g,g,g)
    int nbins, int gdim)
{
    const int atom = blockIdx.x;

    if (!active[atom]) return;                   // uniform: skip DMA + work

    __shared__ float s_row[LDS_ROW_CAP];

    // -------- Stage this atom's radial-density row into LDS via the TDM ----
    // Only wave 0 reaches the tensor op (other waves branch around it); the
    // TDM issues once for that wave regardless of EXEC. D# layout per ISA §8.
    if (threadIdx.x < 32) {
        unsigned long long ga =
            (unsigned long long)(const void*)(rad + (size_t)atom * (size_t)nbins);
        unsigned int lds_addr =
            (unsigned int)(unsigned long long)(void*)&s_row[0];

        v4u g0;
        g0.x = 1u;                                   // count=1 (valid), user mode
        g0.y = lds_addr;                             // lds_addr [63:32]
        g0.z = (unsigned int)ga;                     // global_addr[31:0]
        g0.w = ((unsigned int)(ga >> 32) & 0x01FFFFFFu) | (2u << 30); // [56:32]|type=2

        v8i g1;
        g1[0] = 0x00020000;                          // data_size=2 (4 bytes/elem)
        g1[1] = (nbins & 0xFFFF) << 16;              // tensor_dim0 lo16 @ [63:48]
        g1[2] = (int)(((unsigned)nbins >> 16) | (1u << 16)); // dim0 hi16 | dim1=1
        g1[3] = (nbins & 0xFFFF) << 16;              // tile_dim0 = nbins @ [127:112]
        g1[4] = 1;                                   // tile_dim1=1, tile_dim2=0
        g1[5] = nbins;                               // tensor_dim0_stride lo32
        g1[6] = 0;
        g1[7] = 0;                                   // tensor_dim1_stride = 0

        v4i g2; g2[0] = 1; g2[1] = 1; g2[2] = 0; g2[3] = 0;        // dim2=1, dim3=1
        v4i g3; g3[0] = 0; g3[1] = 0x00010000; g3[2] = 0; g3[3] = 0; // dim4=1

#if __clang_major__ >= 23
        v8i gpad = {0, 0, 0, 0, 0, 0, 0, 0};
        __builtin_amdgcn_tensor_load_to_lds(g0, g1, g2, g3, gpad, 0);
#else
        __builtin_amdgcn_tensor_load_to_lds(g0, g1, g2, g3, 0);
#endif
        __builtin_amdgcn_s_wait_tensorcnt(0);        // TENSORcnt -> 0
    }
    __syncthreads();                                 // publish LDS row to all waves

    // -------- Uniform per-atom scalars (block-uniform -> scalar loads) -----
    const float cx = coords[3 * atom + 0];
    const float cy = coords[3 * atom + 1];
    const float cz = coords[3 * atom + 2];
    const float lx = lmax[0], ly = lmax[1], lz = lmax[2];

    const int b0 = (int)ceilf(cx - lx);
    const int b1 = (int)ceilf(cy - ly);
    const int b2 = (int)ceilf(cz - lz);
    const int e0 = (int)floorf(cx + lx);
    const int e1 = (int)floorf(cy + ly);
    const int e2 = (int)floorf(cz + lz);

    // block-level periodic wrap of the box origin (SALU, once):
    // per-voxel wrap then needs only add + conditional subtract (BOX <= gdim).
    int wb0 = b0 % gdim; if (wb0 < 0) wb0 += gdim;
    int wb1 = b1 % gdim; if (wb1 < 0) wb1 += gdim;
    int wb2 = b2 % gdim; if (wb2 < 0) wb2 += gdim;

    const float g00 = G[0], g01 = G[1], g02 = G[2];
    const float g10 = G[3], g11 = G[4], g12 = G[5];
    const float g20 = G[6], g21 = G[7], g22 = G[8];

    const float rstep = rstep_p[0];
    const float rmaxv = rmax_p[0];
    const float r2max = rmaxv * rmaxv;
    const float ov    = occ[atom];

    // -------- 2197 voxels, 256 threads: innermost offset -> grid dim 2 ----
    for (int v = (int)threadIdx.x; v < NVOX; v += TPB) {
        const int o2 = v % BOX;
        const int t  = v / BOX;
        const int o1 = t % BOX;
        const int o0 = t / BOX;

        const int P0 = b0 + o0;
        const int P1 = b1 + o1;
        const int P2 = b2 + o2;
        if (P0 > e0 || P1 > e1 || P2 > e2) continue; // inbox mask

        const float r0 = (float)P0 - cx;
        const float r1 = (float)P1 - cy;
        const float r2 = (float)P2 - cz;

        const float c0 = g00 * r0 + g01 * r1 + g02 * r2;
        const float c1 = g10 * r0 + g11 * r1 + g12 * r2;
        const float c2 = g20 * r0 + g21 * r1 + g22 * r2;
        const float d2 = c0 * c0 + c1 * c1 + c2 * c2;
        if (d2 > r2max) continue;                    // sphere mask

        const float dist = sqrtf(fmaxf(d2, 1e-12f));
        int idx = (int)(dist / rstep);               // truncation, as reference
        idx = idx < 0 ? 0 : (idx > nbins - 1 ? nbins - 1 : idx);

        const float val = s_row[idx] * ov;           // LDS gather (TDM-staged)

        int m0 = wb0 + o0; m0 = (m0 >= gdim) ? m0 - gdim : m0;  // cheap wrap
        int m1 = wb1 + o1; m1 = (m1 >= gdim) ? m1 - gdim : m1;
        int m2 = wb2 + o2; m2 = (m2 >= gdim) ? m2 - gdim : m2;

        const unsigned int off =
            ((unsigned int)m0 * (unsigned int)gdim + (unsigned int)m1)
                * (unsigned int)gdim + (unsigned int)m2;  // < 8M: 32-bit math
        atomicAdd(out + off, val);
    }
}

extern "C" void kernel_launch(void* const* d_in, const int* in_sizes, int n_in,
                              void* d_out, int out_size, void* d_ws, size_t ws_size,
                              hipStream_t stream) {
    const float*         coords = (const float*)d_in[0];
    const unsigned char* active = (const unsigned char*)d_in[1]; // jax bool = 1 byte
    const float*         occ    = (const float*)d_in[2];
    const float*         lmax   = (const float*)d_in[3];
    const float*         rad    = (const float*)d_in[4];
    const float*         rstep  = (const float*)d_in[5];
    const float*         rmax   = (const float*)d_in[6];
    const float*         G      = (const float*)d_in[7];
    const float*         out_in = (const float*)d_in[8];
    float*               out    = (float*)d_out;

    const int n_atoms = in_sizes[0] / 3;
    const int nbins   = (n_atoms > 0) ? in_sizes[4] / n_atoms : 0;

    // cubic grid: find gdim with gdim^3 == out_size
    int gdim = 1;
    while ((long long)gdim * gdim * gdim < (long long)out_size) ++gdim;

    // out starts as the input grid; accumulate on top of it
    hipMemcpyAsync(out, out_in, (size_t)out_size * sizeof(float),
                   hipMemcpyDeviceToDevice, stream);

    dim3 grid(n_atoms), block(TPB);
    hipLaunchKernelGGL(atom_splat_kernel, grid, block, 0, stream,
                       coords, active, occ, lmax, rad, rstep, rmax, G, out,
                       nbins, gdim);
}